// MS_33767032881693
// MI455X (gfx1250) — compile-verified
//
#include <hip/hip_runtime.h>

// Problem constants (match reference).
#define D        64
#define N        4096          // W*H = 64*64
#define BATCH    2
#define ITERS    5
#define MT       64            // K-matrix columns per workgroup
#define NT       64            // n-block (rows of K tile) per pass
#define LS       76            // LDS row stride (12 mod 64: conflict-free patterns)
#define BW       0.1f
#define STEP     0.5f

typedef __attribute__((ext_vector_type(2))) float v2f;
typedef __attribute__((ext_vector_type(8))) float v8f;

// D = A(16x4, f32) * B(4x16, f32) + C(16x16, f32)  -> v_wmma_f32_16x16x4_f32
__device__ __forceinline__ v8f wmma4(v2f a, v2f b, v8f c) {
    return __builtin_amdgcn_wmma_f32_16x16x4_f32(false, a, false, b,
                                                 (short)0, c, false, false);
}

// out[b, 0, :, :] = x_in[b]
__global__ void MS_copy_init(const float* __restrict__ x,
                             float* __restrict__ out, int total) {
    int i = blockIdx.x * blockDim.x + threadIdx.x;
    if (i < total) {
        int b = i / (D * N);
        int r = i - b * (D * N);
        out[(size_t)b * (ITERS + 1) * D * N + r] = x[i];
    }
}

// One mean-shift iteration: reads slot t, writes slot t+1.
// Block = 256 threads (8 wave32). blockIdx.x = m-tile, blockIdx.y = image.
__global__ void __launch_bounds__(256)
MS_iter_kernel(float* __restrict__ out, int t) {
    const int b = blockIdx.y;
    const float* __restrict__ yin = out + ((size_t)b * (ITERS + 1) + t) * D * N;
    float* __restrict__ yo        = out + ((size_t)b * (ITERS + 1) + t + 1) * D * N;
    const int m0 = blockIdx.x * MT;

    __shared__ float Ya [D ][LS];     // Y[:, nb:nb+64], d-major (restaged per n-block)
    __shared__ float Ybt[MT][LS];     // Y[:, m0:m0+64], TRANSPOSED: [m][d] (staged once)
    __shared__ float Klt[MT][LS];     // exp'd gram tile, TRANSPOSED: [m][n]
    __shared__ float colsum[MT];      // accumulated over all n-blocks

    const int tid  = threadIdx.x;
    const int lane = tid & 31;
    const int wave = tid >> 5;
    const int l16  = lane & 15;
    const int hi   = lane >> 4;       // 0: lanes 0-15, 1: lanes 16-31
    const int koff = hi * 2;          // fragment K split per ISA layout

    // Cooperative tile staging: thread -> row tid>>2, 16 cols at (tid&3)*16.
    const int sr = tid >> 2;          // 0..63 (d row)
    const int sc = (tid & 3) * 16;    // 0,16,32,48

    if (tid < MT) colsum[tid] = 0.0f;

    // Stage loop-invariant Ym tile, transposed into [m][d] (once per block).
    #pragma unroll
    for (int j = 0; j < 16; j += 4) {
        const float4 v = *(const float4*)&yin[(size_t)sr * N + m0 + sc + j];
        Ybt[sc + j + 0][sr] = v.x;
        Ybt[sc + j + 1][sr] = v.y;
        Ybt[sc + j + 2][sr] = v.z;
        Ybt[sc + j + 3][sr] = v.w;
    }
    __syncthreads();

    // Wave -> two 16x16 tiles sharing a tile-row (shared A fragment).
    const int tr  = wave >> 1;        // tile row 0..3 (n-rows S1, d-rows S2)
    const int tc0 = (wave & 1) * 2;   // tile cols tc0, tc0+1 (of 4)
    const int c0  = tc0 * 16 + l16;   // this lane's column (tile 0) in [0,64)
    const int c1  = c0 + 16;          // this lane's column (tile 1)

    v8f z0 = {}; v8f z1 = {};         // persistent Z accumulators (D x MT slab)

    for (int nb = 0; nb < N; nb += NT) {
        // ---- stage Yn tile, d-major (stage-1 A strided, stage-2 A contig) ----
        #pragma unroll
        for (int j = 0; j < 16; j += 4) {
            const float4 v = *(const float4*)&yin[(size_t)sr * N + nb + sc + j];
            *(float4*)&Ya[sr][sc + j] = v;
        }
        __syncthreads();

        // ---------- Stage 1: Ktile = exp(BW * Yn^T * Ym), 64x64 ----------
        v8f g0 = {}; v8f g1 = {};
        {
            const int nl = tr * 16 + l16;       // A row (M) = local n index
            #pragma unroll
            for (int k = 0; k < D; k += 4) {
                v2f af, b0f, b1f;
                af.x = Ya[k + koff    ][nl];    // -> one ds_load_2addr pair
                af.y = Ya[k + koff + 1][nl];
                b0f  = *(const v2f*)&Ybt[c0][k + koff];   // ds_load_b64
                b1f  = *(const v2f*)&Ybt[c1][k + koff];   // ds_load_b64
                g0 = wmma4(af, b0f, g0);
                g1 = wmma4(af, b1f, g1);
            }
        }
        // exp -> transposed K tile (each lane owns 8 consecutive n of one m row),
        // plus per-column partial sums folded in-register.
        {
            float4 e0a, e0b, e1a, e1b;
            float* p0 = (float*)&e0a;           // e0a.x..w, then e0b via p0[4..7]? no:
            // compute explicitly to keep vector stores simple
            float cs0, cs1;
            {
                float t0[8], t1[8];
                #pragma unroll
                for (int v = 0; v < 8; ++v) {
                    t0[v] = __expf(BW * g0[v]);
                    t1[v] = __expf(BW * g1[v]);
                }
                cs0 = (((t0[0] + t0[1]) + (t0[2] + t0[3])) +
                       ((t0[4] + t0[5]) + (t0[6] + t0[7])));
                cs1 = (((t1[0] + t1[1]) + (t1[2] + t1[3])) +
                       ((t1[4] + t1[5]) + (t1[6] + t1[7])));
                e0a = make_float4(t0[0], t0[1], t0[2], t0[3]);
                e0b = make_float4(t0[4], t0[5], t0[6], t0[7]);
                e1a = make_float4(t1[0], t1[1], t1[2], t1[3]);
                e1b = make_float4(t1[4], t1[5], t1[6], t1[7]);
            }
            const int r0 = tr * 16 + hi * 8;    // first local-n row of this lane
            *(float4*)&Klt[c0][r0    ] = e0a;   // ds_store_b128 x4
            *(float4*)&Klt[c0][r0 + 4] = e0b;
            *(float4*)&Klt[c1][r0    ] = e1a;
            *(float4*)&Klt[c1][r0 + 4] = e1b;
            atomicAdd(&colsum[c0], cs0);        // ds_add_f32
            atomicAdd(&colsum[c1], cs1);
        }
        __syncthreads();

        // ---------- Stage 2: Z += Y(:, nb:nb+NT) * Ktile ----------
        {
            const int drow = tr * 16 + l16;     // A row (M) = d index
            #pragma unroll
            for (int k = 0; k < NT; k += 4) {
                v2f af, b0f, b1f;
                af  = *(const v2f*)&Ya [drow][k + koff];  // ds_load_b64
                b0f = *(const v2f*)&Klt[c0  ][k + koff];  // ds_load_b64
                b1f = *(const v2f*)&Klt[c1  ][k + koff];  // ds_load_b64
                z0 = wmma4(af, b0f, z0);
                z1 = wmma4(af, b1f, z1);
            }
        }
        __syncthreads();                        // before Ya/Klt are overwritten
    }

    // ---------- finalize: y' = STEP * Z / colsum + (1-STEP) * y ----------
    {
        const int mA = m0 + c0;
        const int mB = m0 + c1;
        const float r0 = 1.0f / colsum[c0];
        const float r1 = 1.0f / colsum[c1];
        #pragma unroll
        for (int v = 0; v < 8; ++v) {
            int d = tr * 16 + v + hi * 8;
            size_t i0 = (size_t)d * N + mA;
            size_t i1 = (size_t)d * N + mB;
            yo[i0] = STEP * z0[v] * r0 + (1.0f - STEP) * yin[i0];
            yo[i1] = STEP * z1[v] * r1 + (1.0f - STEP) * yin[i1];
        }
    }
}

extern "C" void kernel_launch(void* const* d_in, const int* in_sizes, int n_in,
                              void* d_out, int out_size, void* d_ws, size_t ws_size,
                              hipStream_t stream) {
    const float* x = (const float*)d_in[0];
    float* out = (float*)d_out;

    const int total = BATCH * D * N;
    MS_copy_init<<<(total + 255) / 256, 256, 0, stream>>>(x, out, total);

    dim3 grid(N / MT, BATCH);
    for (int t = 0; t < ITERS; ++t) {
        MS_iter_kernel<<<grid, 256, 0, stream>>>(out, t);
    }
}